// RmseSample_7301444403962
// MI455X (gfx1250) — compile-verified
//
#include <hip/hip_runtime.h>
#include <hip/hip_bf16.h>

typedef __attribute__((ext_vector_type(2))) float v2f;
typedef __attribute__((ext_vector_type(4))) float v4f;
typedef __attribute__((ext_vector_type(8))) float v8f;

#define NT    2000
#define NS    10000
#define NY    3
#define NCOL  (NS * NY)      /* 30000 columns (s,y) */
#define NCOL4 (NCOL / 4)     /* 7500 float4 columns */
#define TSLAB 50
#define NSLAB ((NT + TSLAB - 1) / TSLAB)  /* 40 */
#define BLK   256

// ---------------- Pass 1: bandwidth-bound masked partial sums ----------------
// Thread owns 4 consecutive (s,y) columns; loops over a slab of t rows.
// Non-temporal b128 loads (480MB single-pass stream > 192MB L2, so NT keeps
// the 240KB atomic accumulator region resident); register accumulation;
// f32 atomics to L2.
__global__ __launch_bounds__(BLK) void rmse_partial(
    const float* __restrict__ out, const float* __restrict__ tgt,
    float* __restrict__ sums, float* __restrict__ cnts)
{
    int c4 = blockIdx.x * BLK + threadIdx.x;
    if (c4 >= NCOL4) return;
    int col = c4 * 4;
    int t0 = blockIdx.y * TSLAB;
    int t1 = t0 + TSLAB; if (t1 > NT) t1 = NT;

    const v4f* po = (const v4f*)(out + (size_t)t0 * NCOL + col);
    const v4f* pt = (const v4f*)(tgt + (size_t)t0 * NCOL + col);

    v4f s = {0.f, 0.f, 0.f, 0.f};
    v4f n = {0.f, 0.f, 0.f, 0.f};

#pragma unroll 2
    for (int t = t0; t < t1; ++t) {
        v4f o = __builtin_nontemporal_load(po);
        v4f g = __builtin_nontemporal_load(pt);
        {
            float d = o.x - g.x; bool v = !__builtin_isnan(d);
            float dd = v ? d : 0.f; s.x += dd * dd; n.x += v ? 1.f : 0.f;
        }
        {
            float d = o.y - g.y; bool v = !__builtin_isnan(d);
            float dd = v ? d : 0.f; s.y += dd * dd; n.y += v ? 1.f : 0.f;
        }
        {
            float d = o.z - g.z; bool v = !__builtin_isnan(d);
            float dd = v ? d : 0.f; s.z += dd * dd; n.z += v ? 1.f : 0.f;
        }
        {
            float d = o.w - g.w; bool v = !__builtin_isnan(d);
            float dd = v ? d : 0.f; s.w += dd * dd; n.w += v ? 1.f : 0.f;
        }
        po += NCOL / 4; pt += NCOL / 4;
    }

    atomicAdd(&sums[col + 0], s.x);
    atomicAdd(&sums[col + 1], s.y);
    atomicAdd(&sums[col + 2], s.z);
    atomicAdd(&sums[col + 3], s.w);
    atomicAdd(&cnts[col + 0], n.x);
    atomicAdd(&cnts[col + 1], n.y);
    atomicAdd(&cnts[col + 2], n.z);
    atomicAdd(&cnts[col + 3], n.w);
}

// ---------------- Pass 2: finalize (single block, WMMA tree reduce) ----------
// Each thread folds NS/256 samples into per-channel partials {sum(mean), cs}.
// Cross-thread reduction: v_wmma_f32_16x16x4_f32 with all-ones A matrix.
// With A == ones, C[m][n] = sum_k B[k][n] for every m, so 4 chained WMMAs
// fold 4*64 = 256 lane-partials into 16 column sums (layout-permutation
// invariant because ones-A sums all K rows symmetrically).
__global__ __launch_bounds__(BLK) void rmse_finalize(
    const float* __restrict__ sums, const float* __restrict__ cnts,
    float* __restrict__ outp)
{
    __shared__ float pS[3 * BLK];   // per-thread per-channel sum(mean)
    __shared__ float pC[3 * BLK];   // per-thread per-channel valid count
    __shared__ float red[6 * 16];   // 16 WMMA column-sums per channel/kind

    int tid = threadIdx.x;

    float a0 = 0.f, a1 = 0.f, a2 = 0.f;
    float c0 = 0.f, c1 = 0.f, c2 = 0.f;
    for (int sI = tid; sI < NS; sI += BLK) {
        int base = sI * NY;
        float s0 = sums[base + 0], s1 = sums[base + 1], s2 = sums[base + 2];
        float n0 = cnts[base + 0], n1 = cnts[base + 1], n2 = cnts[base + 2];
        // mean = sum / max(cnt,1); when cnt > 0, max(cnt,1) == cnt
        a0 += (n0 > 0.f) ? s0 / n0 : 0.f;  c0 += (n0 > 0.f) ? 1.f : 0.f;
        a1 += (n1 > 0.f) ? s1 / n1 : 0.f;  c1 += (n1 > 0.f) ? 1.f : 0.f;
        a2 += (n2 > 0.f) ? s2 / n2 : 0.f;  c2 += (n2 > 0.f) ? 1.f : 0.f;
    }
    pS[0 * BLK + tid] = a0;  pS[1 * BLK + tid] = a1;  pS[2 * BLK + tid] = a2;
    pC[0 * BLK + tid] = c0;  pC[1 * BLK + tid] = c1;  pC[2 * BLK + tid] = c2;
    __syncthreads();

    int w = tid >> 5;   // wave id (wave32)
    int l = tid & 31;   // lane id
    if (w < 6) {        // wave-uniform branch: EXEC all-ones inside
        const float* P = (w < 3) ? &pS[w * BLK] : &pC[(w - 3) * BLK];
        v2f a; a[0] = 1.0f; a[1] = 1.0f;   // A = ones(16x4)
        v8f c = {};                         // C accumulator = 0
#pragma unroll
        for (int j = 0; j < 4; ++j) {
            v2f b;
            b[0] = P[l + 64 * j];
            b[1] = P[l + 32 + 64 * j];
            c = __builtin_amdgcn_wmma_f32_16x16x4_f32(
                    false, a, false, b, (short)0, c, false, false);
        }
        // C row 0 (VGPR 0): lanes 0..15 hold a permutation of the 16 col sums
        if (l < 16) red[w * 16 + l] = c[0];
    }
    __syncthreads();

    if (tid == 0) {
        float res = 0.f;
        for (int y = 0; y < 3; ++y) {
            float S = 0.f, C = 0.f;
            for (int i = 0; i < 16; ++i) {
                S += red[y * 16 + i];
                C += red[(y + 3) * 16 + i];
            }
            res += (C > 0.f) ? (S / C) : 0.f;
        }
        outp[0] = res / (float)NY;
    }
}

extern "C" void kernel_launch(void* const* d_in, const int* in_sizes, int n_in,
                              void* d_out, int out_size, void* d_ws, size_t ws_size,
                              hipStream_t stream) {
    const float* out_t = (const float*)d_in[0];  // "output" [NT,NS,NY] f32
    const float* tgt_t = (const float*)d_in[1];  // "target" [NT,NS,NY] f32
    float* sums = (float*)d_ws;                  // [NCOL]
    float* cnts = sums + NCOL;                   // [NCOL]

    // zero accumulators every launch (graph-capture-safe memset node)
    (void)hipMemsetAsync(d_ws, 0, 2 * NCOL * sizeof(float), stream);

    dim3 grid1((NCOL4 + BLK - 1) / BLK, NSLAB);
    rmse_partial<<<grid1, BLK, 0, stream>>>(out_t, tgt_t, sums, cnts);
    rmse_finalize<<<1, BLK, 0, stream>>>(sums, cnts, (float*)d_out);
}